// Scatter_45337674776576
// MI455X (gfx1250) — compile-verified
//
#include <hip/hip_runtime.h>

typedef __attribute__((ext_vector_type(2))) float v2f;
typedef __attribute__((ext_vector_type(8))) float v8f;

#define NEG_SLOPE 0.2f
#define BN_EPS 1e-5f

// ---------------- helpers ----------------
__device__ __forceinline__ unsigned enc_ord(float f) {
  unsigned u = __float_as_uint(f);
  return u ^ (unsigned)(((int)u >> 31) | 0x80000000);
}
__device__ __forceinline__ float dec_ord(unsigned e) {
  return (e & 0x80000000u) ? __uint_as_float(e ^ 0x80000000u) : __uint_as_float(~e);
}
// branchless exact ELU: max(x,0) + (exp(min(x,0)) - 1); no EXEC manipulation
__device__ __forceinline__ float elu_nb(float x) {
  return fmaxf(x, 0.0f) + (__expf(fminf(x, 0.0f)) - 1.0f);
}

__global__ void zero_f32(float* __restrict__ p, int n) {
  int t = blockIdx.x * blockDim.x + threadIdx.x;
  if (t < n) p[t] = 0.0f;
}

// ---------------- scatter pass: out[dst] += ea * X[src], C=128 ----------------
__global__ void scatter_edges(const int* __restrict__ src, const int* __restrict__ dst,
                              const float* __restrict__ ea, const float* __restrict__ X,
                              float* __restrict__ out, int E) {
  int t = blockIdx.x * blockDim.x + threadIdx.x;
  if (t >= E * 32) return;
  int e = t >> 5;     // edge
  int q = t & 31;     // float4 chunk (32 * 4 = 128 channels)
  int s = src[e];
  int d = dst[e];
  float w = ea[e];
  float4 v = ((const float4*)(X + (size_t)s * 128))[q];
  float* o = out + (size_t)d * 128 + (size_t)q * 4;
  atomicAdd(o + 0, w * v.x);
  atomicAdd(o + 1, w * v.y);
  atomicAdd(o + 2, w * v.z);
  atomicAdd(o + 3, w * v.w);
}

// ---------------- fp32 WMMA GEMM: D = opA(A) @ B (+bias) ----------------
// A: [Nrow,K] row-major, B: [K,NCOL] row-major, D: [Nrow,NCOL]
// ELU: A element -> elu(A + addA[k]) (used for the MLP: elu(g + gb))
// One wave computes one 16x16 tile via V_WMMA_F32_16X16X4_F32, stepping K by 4.
// K / NCOL are compile-time: all loads become base + immediate offset.
template <int K, int NCOL, bool ELU>
__global__ void gemm_wmma_f32(const float* __restrict__ A, const float* __restrict__ B,
                              const float* __restrict__ addA,
                              const float* __restrict__ bias,
                              float* __restrict__ D, int Nrow) {
  int wave = (blockIdx.x * blockDim.x + threadIdx.x) >> 5;
  int lane = threadIdx.x & 31;
  constexpr int tiles_n = NCOL >> 4;
  int tiles_m = Nrow >> 4;
  if (wave >= tiles_m * tiles_n) return;   // wave-uniform -> EXEC stays all-1
  int row0 = (wave / tiles_n) * 16;
  int col0 = (wave % tiles_n) * 16;
  int half = lane >> 4;    // ISA A-layout: lanes 0-15 hold K{0,1}, lanes 16-31 hold K{2,3}
  int ml = lane & 15;
  const v2f* arow2 = (const v2f*)(A + (size_t)(row0 + ml) * K);
  const v2f* add2  = ELU ? (const v2f*)addA : nullptr;
  int c = col0 + ml;
  const float* bcol = B + c;
  v8f acc = {};
#pragma unroll 8
  for (int kk = 0; kk < (K >> 2); ++kk) {
    const int k0 = kk * 4 + 2 * half;
    v2f a = arow2[2 * kk + half];
    if (ELU) {
      v2f ad = add2[2 * kk + half];
      a.x = elu_nb(a.x + ad.x);
      a.y = elu_nb(a.y + ad.y);
    }
    v2f b;
    b.x = bcol[(size_t)k0 * NCOL];
    b.y = bcol[(size_t)(k0 + 1) * NCOL];
    acc = __builtin_amdgcn_wmma_f32_16x16x4_f32(false, a, false, b, (short)0, acc,
                                                false, false);
  }
  float bv = bias ? bias[c] : 0.0f;
#pragma unroll
  for (int j = 0; j < 8; ++j) {  // C/D layout: VGPR j -> rows j (lanes 0-15) / j+8 (16-31)
    D[(size_t)(row0 + j + 8 * half) * NCOL + c] = acc[j] + bv;
  }
}

// ---------------- GAT: per-node attention logits ----------------
// AS[n,h] = sum_c H[n,h*128+c]*as_w[h,c];  AD likewise. One wave per node.
__global__ void gat_alpha(const float* __restrict__ H, const float* __restrict__ as_w,
                          const float* __restrict__ ad_w, float* __restrict__ AS,
                          float* __restrict__ AD, int N) {
  int n = blockIdx.x * (blockDim.x >> 5) + (threadIdx.x >> 5);
  int lane = threadIdx.x & 31;
  if (n >= N) return;
  const float* hrow = H + (size_t)n * 256;
#pragma unroll
  for (int head = 0; head < 2; ++head) {
    float s = 0.0f, d = 0.0f;
#pragma unroll
    for (int j = lane; j < 128; j += 32) {
      float hv = hrow[head * 128 + j];
      s += hv * as_w[head * 128 + j];
      d += hv * ad_w[head * 128 + j];
    }
    for (int off = 16; off > 0; off >>= 1) {
      s += __shfl_down(s, off, 32);
      d += __shfl_down(d, off, 32);
    }
    if (lane == 0) { AS[n * 2 + head] = s; AD[n * 2 + head] = d; }
  }
}

// ---------------- GAT: segment max of leaky_relu logits ----------------
__global__ void gat_max(const int* __restrict__ ei, int EO, int N,
                        const float* __restrict__ AS, const float* __restrict__ AD,
                        unsigned* __restrict__ M) {
  int e = blockIdx.x * blockDim.x + threadIdx.x;
  if (e >= EO + N) return;
  int s = (e < EO) ? ei[e] : (e - EO);
  int d = (e < EO) ? ei[EO + e] : (e - EO);
  float2 av = *(const float2*)(AS + (size_t)s * 2);
  float2 dv = *(const float2*)(AD + (size_t)d * 2);
  float e0 = av.x + dv.x, e1 = av.y + dv.y;
  e0 = e0 > 0.0f ? e0 : NEG_SLOPE * e0;
  e1 = e1 > 0.0f ? e1 : NEG_SLOPE * e1;
  atomicMax(&M[(size_t)d * 2 + 0], enc_ord(e0));
  atomicMax(&M[(size_t)d * 2 + 1], enc_ord(e1));
}

// ---------------- GAT: exp + segment sum ----------------
__global__ void gat_exp(const int* __restrict__ ei, int EO, int N,
                        const float* __restrict__ AS, const float* __restrict__ AD,
                        const unsigned* __restrict__ M, float* __restrict__ EXPV,
                        float* __restrict__ DEN) {
  int e = blockIdx.x * blockDim.x + threadIdx.x;
  if (e >= EO + N) return;
  int s = (e < EO) ? ei[e] : (e - EO);
  int d = (e < EO) ? ei[EO + e] : (e - EO);
  float2 av = *(const float2*)(AS + (size_t)s * 2);
  float2 dv = *(const float2*)(AD + (size_t)d * 2);
  float e0 = av.x + dv.x, e1 = av.y + dv.y;
  e0 = e0 > 0.0f ? e0 : NEG_SLOPE * e0;
  e1 = e1 > 0.0f ? e1 : NEG_SLOPE * e1;
  float x0 = __expf(e0 - dec_ord(M[(size_t)d * 2 + 0]));
  float x1 = __expf(e1 - dec_ord(M[(size_t)d * 2 + 1]));
  float2 xv; xv.x = x0; xv.y = x1;
  ((float2*)EXPV)[e] = xv;
  atomicAdd(&DEN[(size_t)d * 2 + 0], x0);
  atomicAdd(&DEN[(size_t)d * 2 + 1], x1);
}

// ---------------- GAT: weighted aggregation G[dst] += alpha * H[src] ----------------
__global__ void gat_agg(const int* __restrict__ ei, int EO, int N,
                        const float* __restrict__ EXPV, const float* __restrict__ DEN,
                        const float* __restrict__ H, float* __restrict__ G) {
  int t = blockIdx.x * blockDim.x + threadIdx.x;
  int total = (EO + N) * 64;   // 64 float4 chunks over 256 channels
  if (t >= total) return;
  int e = t >> 6;
  int q = t & 63;
  int s = (e < EO) ? ei[e] : (e - EO);
  int d = (e < EO) ? ei[EO + e] : (e - EO);
  int head = q >> 5;
  float w = EXPV[(size_t)e * 2 + head] / (DEN[(size_t)d * 2 + head] + 1e-16f);
  float4 v = ((const float4*)(H + (size_t)s * 256))[q];
  float* o = G + (size_t)d * 256 + (size_t)q * 4;
  atomicAdd(o + 0, w * v.x);
  atomicAdd(o + 1, w * v.y);
  atomicAdd(o + 2, w * v.z);
  atomicAdd(o + 3, w * v.w);
}

// ---------------- BatchNorm stats (sum, sumsq per channel of 128) ----------------
__global__ void bn_stats(const float* __restrict__ Z, float* __restrict__ stats, int N) {
  int c = threadIdx.x;            // blockDim.x == 128
  int r0 = blockIdx.x * 64;
  int r1 = r0 + 64; if (r1 > N) r1 = N;
  float s = 0.0f, s2 = 0.0f;
  for (int r = r0; r < r1; ++r) {
    float v = Z[(size_t)r * 128 + c];
    s += v; s2 += v * v;
  }
  atomicAdd(&stats[c], s);
  atomicAdd(&stats[128 + c], s2);
}

// ---------------- BatchNorm apply + accumulate into out ----------------
__global__ void bn_out(const float* __restrict__ Z, const float* __restrict__ stats,
                       const float* __restrict__ gamma, const float* __restrict__ beta,
                       float* __restrict__ out, int N) {
  int t = blockIdx.x * blockDim.x + threadIdx.x;
  if (t >= N * 128) return;
  int c = t & 127;
  float inv_n = 1.0f / (float)N;
  float mu = stats[c] * inv_n;
  float var = stats[128 + c] * inv_n - mu * mu;
  out[t] += gamma[c] * (Z[t] - mu) * rsqrtf(var + BN_EPS) + beta[c];
}

// ---------------- launch ----------------
static inline unsigned gridFor(long long total, int block) {
  return (unsigned)((total + block - 1) / block);
}

extern "C" void kernel_launch(void* const* d_in, const int* in_sizes, int n_in,
                              void* d_out, int out_size, void* d_ws, size_t ws_size,
                              hipStream_t stream) {
  const float* x      = (const float*)d_in[0];
  const int*   ei_o   = (const int*)d_in[1];
  const float* Wg     = (const float*)d_in[2];   // [4,128,256]
  const float* att_s  = (const float*)d_in[3];   // [4,2,128]
  const float* att_d  = (const float*)d_in[4];   // [4,2,128]
  const float* gb     = (const float*)d_in[5];   // [4,256]
  const float* lw     = (const float*)d_in[6];   // [4,256,128]
  const float* lb     = (const float*)d_in[7];   // [4,128]
  const float* gamma  = (const float*)d_in[8];   // [4,128]
  const float* beta   = (const float*)d_in[9];   // [4,128]
  const int*   sei[4] = {(const int*)d_in[10], (const int*)d_in[12],
                         (const int*)d_in[14], (const int*)d_in[16]};
  const float* sea[4] = {(const float*)d_in[11], (const float*)d_in[13],
                         (const float*)d_in[15], (const float*)d_in[17]};

  const int N  = in_sizes[0] / 128;
  const int EO = in_sizes[1] / 2;
  const int E  = in_sizes[10] / 2;
  const int EP = EO + N;   // edges + self loops

  // workspace layout (fp32)
  float* w = (float*)d_ws;
  float*    H    = w;  w += (size_t)N * 256;   // GAT hidden h
  float*    G    = w;  w += (size_t)N * 256;   // GAT aggregated output
  float*    XZ   = w;  w += (size_t)N * 128;   // x_i, later z
  float*    XH   = w;  w += (size_t)N * 128;   // intermediate scatter hop
  float*    AS   = w;  w += (size_t)N * 2;
  float*    AD   = w;  w += (size_t)N * 2;
  unsigned* M    = (unsigned*)w; w += (size_t)N * 2;
  float*    DEN  = w;  w += (size_t)N * 2;
  float*    EXPV = w;  w += (size_t)EP * 2;
  float*    ST   = w;  w += 256;

  const int BLK = 256;
  float* out = (float*)d_out;

  zero_f32<<<gridFor((long long)N * 128, BLK), BLK, 0, stream>>>(out, N * 128);

  for (int i = 0; i < 4; ++i) {
    // ---- solo scatter passes -> XZ = x_i ----
    if (i == 0) {
      zero_f32<<<gridFor((long long)N * 128, BLK), BLK, 0, stream>>>(XZ, N * 128);
      scatter_edges<<<gridFor((long long)E * 32, BLK), BLK, 0, stream>>>(
          sei[0], sei[0] + E, sea[0], x, XZ, E);
    } else {
      zero_f32<<<gridFor((long long)N * 128, BLK), BLK, 0, stream>>>(XH, N * 128);
      scatter_edges<<<gridFor((long long)E * 32, BLK), BLK, 0, stream>>>(
          sei[i], sei[i] + E, sea[i], x, XH, E);
      zero_f32<<<gridFor((long long)N * 128, BLK), BLK, 0, stream>>>(XZ, N * 128);
      scatter_edges<<<gridFor((long long)E * 32, BLK), BLK, 0, stream>>>(
          sei[0], sei[0] + E, sea[0], XH, XZ, E);
    }

    // ---- h = x_i @ Wg[i]  (fp32 WMMA, K=128, NCOL=256) ----
    {
      long long tiles = (long long)(N / 16) * (256 / 16);
      gemm_wmma_f32<128, 256, false><<<gridFor(tiles * 32, BLK), BLK, 0, stream>>>(
          XZ, Wg + (size_t)i * 128 * 256, nullptr, nullptr, H, N);
    }

    // ---- attention logits per node ----
    gat_alpha<<<gridFor(((long long)N) * 32, BLK), BLK, 0, stream>>>(
        H, att_s + (size_t)i * 256, att_d + (size_t)i * 256, AS, AD, N);

    // ---- segment softmax over dst ----
    zero_f32<<<gridFor((long long)N * 2, BLK), BLK, 0, stream>>>((float*)M, N * 2);
    gat_max<<<gridFor((long long)EP, BLK), BLK, 0, stream>>>(ei_o, EO, N, AS, AD, M);
    zero_f32<<<gridFor((long long)N * 2, BLK), BLK, 0, stream>>>(DEN, N * 2);
    gat_exp<<<gridFor((long long)EP, BLK), BLK, 0, stream>>>(ei_o, EO, N, AS, AD, M,
                                                             EXPV, DEN);

    // ---- weighted aggregation ----
    zero_f32<<<gridFor((long long)N * 256, BLK), BLK, 0, stream>>>(G, N * 256);
    gat_agg<<<gridFor((long long)EP * 64, BLK), BLK, 0, stream>>>(ei_o, EO, N, EXPV,
                                                                  DEN, H, G);

    // ---- z = elu(G + gb[i]) @ lw[i] + lb[i]  (fp32 WMMA, K=256, NCOL=128) ----
    {
      long long tiles = (long long)(N / 16) * (128 / 16);
      gemm_wmma_f32<256, 128, true><<<gridFor(tiles * 32, BLK), BLK, 0, stream>>>(
          G, lw + (size_t)i * 256 * 128, gb + (size_t)i * 256, lb + (size_t)i * 128,
          XZ, N);
    }

    // ---- batchnorm + accumulate ----
    zero_f32<<<gridFor(256, BLK), BLK, 0, stream>>>(ST, 256);
    bn_stats<<<gridFor(((long long)N + 63) / 64 * 128, 128), 128, 0, stream>>>(XZ, ST, N);
    bn_out<<<gridFor((long long)N * 128, BLK), BLK, 0, stream>>>(
        XZ, ST, gamma + (size_t)i * 128, beta + (size_t)i * 128, out, N);
  }
}